// Tokenizer_89163521065953
// MI455X (gfx1250) — compile-verified
//
#include <hip/hip_runtime.h>

typedef float v2f __attribute__((ext_vector_type(2)));
typedef float v4f __attribute__((ext_vector_type(4)));
typedef float v8f __attribute__((ext_vector_type(8)));

static constexpr int NTOK = 16384;   // 64 * 16 * 16 spatial tokens
static constexpr int E    = 256;     // embed dim (== ZC)
static constexpr int V    = 4096;    // vocab
static constexpr int HW   = 256;     // 16*16 spatial per batch image

// ---------------------------------------------------------------------------
// Kernel 1: pre-quant 1x1 conv:  z[n,e] = sum_c x[n,c]*w_pre[e,c] + b_pre[e]
// x[n,c] = z_enc[b, c, p]  with n = b*256 + p   (strided NCHW read)
// Writes z_ws (row-major n,e) and z_out in NCHW layout.
// ---------------------------------------------------------------------------
__global__ __launch_bounds__(256)
void pre_gemm_kernel(const float* __restrict__ z_enc,
                     const float* __restrict__ w_pre,
                     const float* __restrict__ b_pre,
                     float* __restrict__ z_ws,
                     float* __restrict__ z_out)
{
    const int wave = threadIdx.x >> 5;         // 8 waves / block
    const int lane = threadIdx.x & 31;
    const int half = lane >> 4;                // K-half select (A/B frags)
    const int lr   = lane & 15;                // M (A) / N (B) index

    const int tile = blockIdx.x * 8 + wave;    // 16384 wave-tiles total
    const int mt   = tile >> 4;                // row tile   0..1023
    const int nt   = tile & 15;                // col tile   0..15

    const int n0 = mt * 16 + lr;
    const int bi = n0 >> 8;
    const int pi = n0 & 255;
    const float* xa = z_enc + bi * (E * HW) + pi;      // + c*HW walks channels
    const float* bb = w_pre + (nt * 16 + lr) * E;

    v8f acc = {};
    for (int kk = 0; kk < E; kk += 4) {
        const int k0 = kk + half * 2;
        v2f a, b;
        a.x = xa[(k0    ) * HW];
        a.y = xa[(k0 + 1) * HW];
        b   = *(const v2f*)(bb + k0);
        acc = __builtin_amdgcn_wmma_f32_16x16x4_f32(false, a, false, b,
                                                    (short)0, acc, false, false);
    }

    const int   ecol = nt * 16 + lr;
    const float bias = b_pre[ecol];
    #pragma unroll
    for (int r = 0; r < 8; ++r) {
        const int   n = mt * 16 + half * 8 + r;
        const float v = acc[r] + bias;
        z_ws[n * E + ecol] = v;
        const int b2 = n >> 8, p2 = n & 255;
        z_out[b2 * (E * HW) + ecol * HW + p2] = v;     // NCHW
    }
}

// ---------------------------------------------------------------------------
// Kernel 2: per-codebook-row squared norms  ||emb[v]||^2   (one wave per row)
// ---------------------------------------------------------------------------
__global__ __launch_bounds__(256)
void emb_norm_kernel(const float* __restrict__ emb, float* __restrict__ e_norm)
{
    const int wave = threadIdx.x >> 5;
    const int lane = threadIdx.x & 31;
    const int row  = blockIdx.x * 8 + wave;            // 4096 rows
    const float* er = emb + row * E;
    float s = 0.f;
    for (int c = lane; c < E; c += 32) { const float v = er[c]; s += v * v; }
    #pragma unroll
    for (int m = 16; m >= 1; m >>= 1) s += __shfl_xor(s, m, 32);
    if (lane == 0) e_norm[row] = s;
}

// ---------------------------------------------------------------------------
// Kernel 3: distances + argmin (dominant GEMM: 16384 x 4096, K=256).
// argmin_v(||z||^2 + ||e_v||^2 - 2 z.e_v) == argmin_v(||e_v||^2 - 2 z.e_v)
//
//  * A tile (16 rows x 256) lives in registers: 64 v2f fragments per lane,
//    loaded once -> zero repeated A traffic.
//  * B (codebook) staged through LDS in 32-column tiles shared by all 8
//    waves of the block (8x L2 traffic reduction), double-buffered with
//    global prefetch overlapped with compute.
//  * LDS layout Bs[col][k] with row stride 260: fragment reads hit all 64
//    banks exactly once per wave (conflict-free ds_load_b64).
//  * Two independent accumulator chains per pass (2 column tiles share one
//    A fragment) -> 2x WMMA ILP, one B-fragment load per WMMA.
// ---------------------------------------------------------------------------
__global__ __launch_bounds__(256)
void argmin_kernel(const float* __restrict__ z_ws,
                   const float* __restrict__ emb,
                   const float* __restrict__ e_norm,
                   int*   __restrict__ tok_ws,
                   float* __restrict__ tok_out)
{
    constexpr int BST = 260;                 // LDS row stride (floats)
    __shared__ float Bs[2][32 * BST];        // 2 x 33.3 KB double buffer

    const int tid  = threadIdx.x;
    const int wave = tid >> 5;
    const int lane = tid & 31;
    const int half = lane >> 4;
    const int lr   = lane & 15;
    const int mt   = blockIdx.x * 8 + wave;  // row tile 0..1023

    // ---- A tile in registers: 64 v2f per lane ----
    v2f a[64];
    {
        const float* arow = z_ws + (mt * 16 + lr) * E + half * 2;
        #pragma unroll
        for (int t = 0; t < 64; ++t) a[t] = *(const v2f*)(arow + 4 * t);
    }

    // cooperative B-tile load assignment: thread -> (column, 32-float k chunk)
    const int col_l = tid >> 3;              // 0..31
    const int kc    = (tid & 7) * 32;        // 0,32,...,224

    float minv[8]; int mini[8];
    #pragma unroll
    for (int r = 0; r < 8; ++r) { minv[r] = 3.4e38f; mini[r] = 0x7fffffff; }

    // prefetch + stage tile 0
    v4f pf[8];
    #pragma unroll
    for (int i = 0; i < 8; ++i)
        pf[i] = *(const v4f*)(emb + col_l * E + kc + 4 * i);
    {
        float* s = &Bs[0][col_l * BST + kc];
        #pragma unroll
        for (int i = 0; i < 8; ++i) *(v4f*)(s + 4 * i) = pf[i];
    }
    __syncthreads();

    for (int ct = 0; ct < V / 32; ++ct) {
        const int pb = ct & 1;

        // issue global prefetch of next tile (overlaps the WMMA loop)
        if (ct + 1 < V / 32) {
            #pragma unroll
            for (int i = 0; i < 8; ++i)
                pf[i] = *(const v4f*)(emb + ((ct + 1) * 32 + col_l) * E + kc + 4 * i);
        }

        // ---- 2 column tiles, 2 independent WMMA chains, A from registers ----
        v8f acc0 = {}, acc1 = {};
        const float* bp = &Bs[pb][lr * BST + half * 2];
        #pragma unroll
        for (int t = 0; t < 64; ++t) {
            const v2f b0 = *(const v2f*)(bp + 4 * t);
            const v2f b1 = *(const v2f*)(bp + 16 * BST + 4 * t);
            acc0 = __builtin_amdgcn_wmma_f32_16x16x4_f32(false, a[t], false, b0,
                                                         (short)0, acc0, false, false);
            acc1 = __builtin_amdgcn_wmma_f32_16x16x4_f32(false, a[t], false, b1,
                                                         (short)0, acc1, false, false);
        }

        const int   c0  = ct * 32 + lr;
        const int   c1  = c0 + 16;
        const float en0 = e_norm[c0];
        const float en1 = e_norm[c1];
        #pragma unroll
        for (int r = 0; r < 8; ++r) {
            const float d0 = en0 - 2.0f * acc0[r];
            const float d1 = en1 - 2.0f * acc1[r];
            if (d0 < minv[r] || (d0 == minv[r] && c0 < mini[r])) { minv[r] = d0; mini[r] = c0; }
            if (d1 < minv[r] || (d1 == minv[r] && c1 < mini[r])) { minv[r] = d1; mini[r] = c1; }
        }

        __syncthreads();                           // everyone done with both buffers' readers
        if (ct + 1 < V / 32) {
            float* s = &Bs[(ct + 1) & 1][col_l * BST + kc];
            #pragma unroll
            for (int i = 0; i < 8; ++i) *(v4f*)(s + 4 * i) = pf[i];
            __syncthreads();                       // next tile staged
        }
    }

    // reduce across the 16 lanes of each half-group (halves hold disjoint rows)
    #pragma unroll
    for (int m = 1; m < 16; m <<= 1) {
        #pragma unroll
        for (int r = 0; r < 8; ++r) {
            const float ov = __shfl_xor(minv[r], m, 32);
            const int   oi = __shfl_xor(mini[r], m, 32);
            if (ov < minv[r] || (ov == minv[r] && oi < mini[r])) {
                minv[r] = ov; mini[r] = oi;
            }
        }
    }
    if (lr == 0) {
        #pragma unroll
        for (int r = 0; r < 8; ++r) {
            const int n = mt * 16 + half * 8 + r;  // half 0 -> rows 0..7
            tok_ws[n]  = mini[r];
            tok_out[n] = (float)mini[r];
        }
    }
}

// ---------------------------------------------------------------------------
// Kernel 4: z_q = emb[token]; write row-major staging + NCHW output
// ---------------------------------------------------------------------------
__global__ __launch_bounds__(256)
void gather_kernel(const float* __restrict__ emb,
                   const int*   __restrict__ tok_ws,
                   float* __restrict__ zq_ws,
                   float* __restrict__ zq_out)
{
    const int n = blockIdx.x;              // 16384
    const int e = threadIdx.x;             // 256
    const int t = tok_ws[n];
    const float v = emb[t * E + e];
    zq_ws[n * E + e] = v;
    const int b = n >> 8, p = n & 255;
    zq_out[b * (E * HW) + e * HW + p] = v; // NCHW
}

// ---------------------------------------------------------------------------
// Kernel 5: post-quant 1x1 conv:  rec = z_q @ w_post^T + b_post  (NCHW out)
// ---------------------------------------------------------------------------
__global__ __launch_bounds__(256)
void post_gemm_kernel(const float* __restrict__ zq_ws,
                      const float* __restrict__ w_post,
                      const float* __restrict__ b_post,
                      float* __restrict__ rec_out)
{
    const int wave = threadIdx.x >> 5;
    const int lane = threadIdx.x & 31;
    const int half = lane >> 4;
    const int lr   = lane & 15;

    const int tile = blockIdx.x * 8 + wave;
    const int mt   = tile >> 4;
    const int nt   = tile & 15;

    const float* arow = zq_ws  + (mt * 16 + lr) * E;
    const float* brow = w_post + (nt * 16 + lr) * E;

    v8f acc = {};
    for (int kk = 0; kk < E; kk += 4) {
        const int k0 = kk + half * 2;
        const v2f a = *(const v2f*)(arow + k0);
        const v2f b = *(const v2f*)(brow + k0);
        acc = __builtin_amdgcn_wmma_f32_16x16x4_f32(false, a, false, b,
                                                    (short)0, acc, false, false);
    }

    const int   ocol = nt * 16 + lr;
    const float bias = b_post[ocol];
    #pragma unroll
    for (int r = 0; r < 8; ++r) {
        const int n = mt * 16 + half * 8 + r;
        const int b2 = n >> 8, p2 = n & 255;
        rec_out[b2 * (E * HW) + ocol * HW + p2] = acc[r] + bias;  // NCHW
    }
}

// ---------------------------------------------------------------------------
extern "C" void kernel_launch(void* const* d_in, const int* in_sizes, int n_in,
                              void* d_out, int out_size, void* d_ws, size_t ws_size,
                              hipStream_t stream)
{
    const float* z_enc  = (const float*)d_in[0];
    const float* w_pre  = (const float*)d_in[1];
    const float* b_pre  = (const float*)d_in[2];
    const float* emb    = (const float*)d_in[3];
    const float* w_post = (const float*)d_in[4];
    const float* b_post = (const float*)d_in[5];

    // Output layout: z | z_q | tokens | rec (flattened, return order)
    float* out     = (float*)d_out;
    float* z_out   = out;
    float* zq_out  = z_out  + (size_t)64 * E * HW;   // +4194304
    float* tok_out = zq_out + (size_t)64 * E * HW;   // +4194304
    float* rec_out = tok_out + (size_t)64 * HW;      // +16384

    // Workspace: z_ws(16MB) | zq_ws(16MB) | e_norm(16KB) | tok_ws(64KB)
    char*  ws     = (char*)d_ws;
    float* z_ws   = (float*)ws;
    float* zq_ws  = (float*)(ws + (size_t)NTOK * E * 4);
    float* e_norm = (float*)(ws + (size_t)NTOK * E * 4 * 2);
    int*   tok_ws = (int*)  (ws + (size_t)NTOK * E * 4 * 2 + (size_t)V * 4);

    pre_gemm_kernel <<<dim3(2048), dim3(256), 0, stream>>>(z_enc, w_pre, b_pre, z_ws, z_out);
    emb_norm_kernel <<<dim3(512),  dim3(256), 0, stream>>>(emb, e_norm);
    argmin_kernel   <<<dim3(128),  dim3(256), 0, stream>>>(z_ws, emb, e_norm, tok_ws, tok_out);
    gather_kernel   <<<dim3(NTOK), dim3(256), 0, stream>>>(emb, tok_ws, zq_ws, zq_out);
    post_gemm_kernel<<<dim3(2048), dim3(256), 0, stream>>>(zq_ws, w_post, b_post, rec_out);
}